// Attention_15857019257228
// MI455X (gfx1250) — compile-verified
//
#include <hip/hip_runtime.h>
#include <hip/hip_bf16.h>
#include <math.h>

// ---------------------------------------------------------------------------
// Problem constants (from reference):
//   N_AGENTS=64, BATCH=32, D_IN=704, D_OUT=256
//   M = N_AGENTS*BATCH = 2048 rows, K = 704, N = 256 per head, 6 heads.
// ---------------------------------------------------------------------------
#define NAG   64
#define BATCH 32
#define DIN   704
#define DOUT  256
#define MROWS (NAG * BATCH)          // 2048
#define HEAD_ELEMS (MROWS * DOUT)    // 524288 floats per head

#define BK_PAD   708                 // 704 + 4 pad -> LDS bank stride 4 (mod 64)

typedef __attribute__((ext_vector_type(2))) float v2f;
typedef __attribute__((ext_vector_type(8))) float v8f;

struct GemmArgs {
    const float* E;        // [2048 x 704] encodings
    const float* W[6];     // each [704 x 256]
    const float* B[6];     // each [256]
    float*       out;      // ws: 6 x [2048 x 256] (k1,k2,v1,v2,q1,q2)
};

// ---------------------------------------------------------------------------
// Kernel 1: six linear heads via fp32 WMMA (16x16x4).
//
// Block (256 thr = 8 waves) owns one (head, tileN, 32-M-tile group):
//   - B panel [704 x 16] staged to LDS transposed (Blds[col][k], stride 708),
//     so a lane's B fragment {W[k+2h][col], W[k+2h+1][col]} is ONE ds_load_b64,
//     bank-conflict-free (col*708 mod 64 = col*4).
//   - Each wave computes a 64x16 strip = 4 accumulator tiles, reusing the B
//     fragment across 4 WMMAs per k-step: 4 glb b64 + 1 ds b64 per 4 WMMAs.
//
// Fragment maps (ISA 7.12.2, wave32):
//   A 16x4 f32: lane<16 -> row=lane, K={k,k+1}; lane>=16 -> K={k+2,k+3}.
//   C/D:        VGPR r  -> M = r + 8*(lane>=16), N = lane&15.
// Bias folded into accumulator init (same value for all 8 C VGPRs of a lane).
// ---------------------------------------------------------------------------
__global__ __launch_bounds__(256) void gemm_heads_wmma(GemmArgs args) {
    __shared__ float Blds[16][BK_PAD];           // 45,312 bytes

    const int bid   = blockIdx.x;                // 384 blocks
    const int head  = bid / 64;                  // 6
    const int rem   = bid % 64;
    const int tileN = rem >> 2;                  // 16 N tiles
    const int mblk  = rem & 3;                   // 4 groups of 32 M tiles
    const int wave  = threadIdx.x >> 5;          // 8 waves
    const int lane  = threadIdx.x & 31;

    // ---- Stage B panel [704 x 16] -> Blds[col][k] (transposed) -------------
    // Pair index p: k = p/8, cols {2*(p%8), +1}. 8 threads cover one 64B row.
    {
        const float* __restrict__ Wp = args.W[head] + tileN * 16;
        #pragma unroll
        for (int it = 0; it < 22; ++it) {        // 5632 pairs / 256 threads
            const int p = threadIdx.x + it * 256;
            const int k = p >> 3;
            const int c = (p & 7) * 2;
            v2f w = *(const v2f*)(Wp + (size_t)k * DOUT + c);
            Blds[c][k]     = w.x;                // conflict-free: banks c*4+k
            Blds[c + 1][k] = w.y;
        }
    }
    __syncthreads();

    // ---- Compute: 4 M-tiles (64 rows) per wave -----------------------------
    const int halfsel = lane >> 4;               // K-pair select
    const int l16     = lane & 15;
    const int mtile0  = mblk * 32 + wave * 4;    // first of 4 M tiles
    const int col     = tileN * 16 + l16;

    const float bval = args.B[head][col];
    v8f acc[4];
    #pragma unroll
    for (int t = 0; t < 4; ++t)
        acc[t] = (v8f){ bval, bval, bval, bval, bval, bval, bval, bval };

    const float* __restrict__ A  = args.E
        + (size_t)(mtile0 * 16 + l16) * DIN + halfsel * 2;
    const float* __restrict__ Bl = &Blds[l16][halfsel * 2];

    #pragma unroll 2
    for (int k = 0; k < DIN; k += 4) {
        const v2f b  = *(const v2f*)(Bl + k);                 // ds_load_b64
        const v2f a0 = *(const v2f*)(A + k);                  // global b64 x4
        const v2f a1 = *(const v2f*)(A + 16 * DIN + k);
        const v2f a2 = *(const v2f*)(A + 32 * DIN + k);
        const v2f a3 = *(const v2f*)(A + 48 * DIN + k);
        acc[0] = __builtin_amdgcn_wmma_f32_16x16x4_f32(false, a0, false, b,
                                                       (short)0, acc[0], false, false);
        acc[1] = __builtin_amdgcn_wmma_f32_16x16x4_f32(false, a1, false, b,
                                                       (short)0, acc[1], false, false);
        acc[2] = __builtin_amdgcn_wmma_f32_16x16x4_f32(false, a2, false, b,
                                                       (short)0, acc[2], false, false);
        acc[3] = __builtin_amdgcn_wmma_f32_16x16x4_f32(false, a3, false, b,
                                                       (short)0, acc[3], false, false);
    }

    // ---- Store: tile t, C VGPR r -> (M = t*16 + r + 8*halfsel, N = col) ----
    float* __restrict__ o = args.out + (size_t)head * HEAD_ELEMS
                          + (size_t)(mtile0 * 16) * DOUT + col;
    #pragma unroll
    for (int t = 0; t < 4; ++t)
        #pragma unroll
        for (int r = 0; r < 8; ++r)
            o[(size_t)(t * 16 + r + halfsel * 8) * DOUT] = acc[t][r];
}

// ---------------------------------------------------------------------------
// Kernel 2: per-element attention.
//   out[h*64+i, b, d] = sum_j softmax_j(q[i,b,d]*k[j,b,d]) * v[j,b,d]
// No contraction over d -> VALU + LDS streaming, not a GEMM.
// Block handles one (b, head, d-chunk of 128); k/v tiles (64x128 each) in LDS.
// ---------------------------------------------------------------------------
__global__ __launch_bounds__(128) void attn_softmax(const float* __restrict__ ws,
                                                    float* __restrict__ out) {
    const int dchunk = blockIdx.x & 1;
    const int head   = (blockIdx.x >> 1) & 1;   // 0: (q1,k1,v1)  1: (q2,k2,v2)
    const int b      = blockIdx.x >> 2;         // 0..31
    const int tid    = threadIdx.x;             // 0..127
    const int d      = dchunk * 128 + tid;

    // ws layout: 0=k1, 1=k2, 2=v1, 3=v2, 4=q1, 5=q2
    const float* __restrict__ kk = ws + (size_t)(0 + head) * HEAD_ELEMS;
    const float* __restrict__ vv = ws + (size_t)(2 + head) * HEAD_ELEMS;
    const float* __restrict__ qq = ws + (size_t)(4 + head) * HEAD_ELEMS;

    __shared__ float kl[NAG][128];   // 32 KB
    __shared__ float vl[NAG][128];   // 32 KB

    #pragma unroll 4
    for (int j = 0; j < NAG; ++j) {
        const size_t src = (size_t)(j * BATCH + b) * DOUT + d;
        kl[j][tid] = kk[src];
        vl[j][tid] = vv[src];
    }
    __syncthreads();

    for (int i = 0; i < NAG; ++i) {
        const float qv = qq[(size_t)(i * BATCH + b) * DOUT + d];

        // Pass 1: running max over j (softmax stabilization, matches jax).
        float mx = -3.402823466e+38f;
        #pragma unroll 8
        for (int j = 0; j < NAG; ++j)
            mx = fmaxf(mx, qv * kl[j][tid]);

        // Pass 2: exp-sum and weighted value accumulation.
        float se = 0.0f, accv = 0.0f;
        #pragma unroll 8
        for (int j = 0; j < NAG; ++j) {
            const float e = __expf(qv * kl[j][tid] - mx);
            se   += e;
            accv += e * vl[j][tid];
        }

        out[((size_t)(head * NAG + i) * BATCH + b) * DOUT + d] = accv / se;
    }
}

// ---------------------------------------------------------------------------
// Launch: d_in order = encodings, Wk1,bk1, Wk2,bk2, Wv1,bv1, Wv2,bv2,
//                      Wq1,bq1, Wq2,bq2
// d_ws needs 6 * 2048 * 256 * 4 = 12 MB for the head outputs.
// ---------------------------------------------------------------------------
extern "C" void kernel_launch(void* const* d_in, const int* in_sizes, int n_in,
                              void* d_out, int out_size, void* d_ws, size_t ws_size,
                              hipStream_t stream) {
    (void)in_sizes; (void)n_in; (void)out_size; (void)ws_size;

    GemmArgs ga;
    ga.E = (const float*)d_in[0];
    // ws head order: 0=k1, 1=k2, 2=v1, 3=v2, 4=q1, 5=q2
    ga.W[0] = (const float*)d_in[1];  ga.B[0] = (const float*)d_in[2];   // k1
    ga.W[1] = (const float*)d_in[3];  ga.B[1] = (const float*)d_in[4];   // k2
    ga.W[2] = (const float*)d_in[5];  ga.B[2] = (const float*)d_in[6];   // v1
    ga.W[3] = (const float*)d_in[7];  ga.B[3] = (const float*)d_in[8];   // v2
    ga.W[4] = (const float*)d_in[9];  ga.B[4] = (const float*)d_in[10];  // q1
    ga.W[5] = (const float*)d_in[11]; ga.B[5] = (const float*)d_in[12];  // q2
    ga.out  = (float*)d_ws;

    // 6 heads x 16 N-tiles x 4 M-groups = 384 blocks; each wave does a 64x16
    // strip (4 WMMA tiles), 8 waves per block share one LDS B panel.
    gemm_heads_wmma<<<384, 256, 0, stream>>>(ga);

    // 32 batches x 2 heads x 2 d-chunks = 128 blocks of 128 threads.
    attn_softmax<<<128, 128, 0, stream>>>((const float*)d_ws, (float*)d_out);
}